// Temp_FAM_RelGraphConv_8263517078050
// MI455X (gfx1250) — compile-verified
//
#include <hip/hip_runtime.h>
#include <hip/hip_bf16.h>

typedef float v2f __attribute__((ext_vector_type(2)));
typedef float v8f __attribute__((ext_vector_type(8)));

// ---------------------------------------------------------------------------
// Kernel 1: per-relation projection GEMMs + self-loop GEMM, via
// V_WMMA_F32_16X16X4_F32. One wave computes a 16-row x 64-col output slab.
// rsel in [0, R-1] -> proj[rsel] = feat @ weight[rsel]   (written to d_ws)
// rsel == R        -> out       = feat @ loop_weight     (initializes d_out)
// ---------------------------------------------------------------------------
__global__ void rgcn_proj_wmma(const float* __restrict__ feat,      // [N,64]
                               const float* __restrict__ weight,    // [R,64,64]
                               const float* __restrict__ loop_w,    // [64,64]
                               float* __restrict__ proj,            // [R,N,64]
                               float* __restrict__ out,             // [N,64]
                               int N, int R, int n_mblocks) {
  const int wave  = blockIdx.x * (blockDim.x >> 5) + (threadIdx.x >> 5);
  const int total = n_mblocks * (R + 1);
  if (wave >= total) return;                 // uniform per-wave exit: EXEC stays all-1s

  const int rsel = wave / n_mblocks;
  const int mb   = wave % n_mblocks;
  const int m0   = mb * 16;

  const float* __restrict__ W =
      (rsel == R) ? loop_w : (weight + (size_t)rsel * 64 * 64);
  float* __restrict__ dst =
      (rsel == R) ? out : (proj + (size_t)rsel * (size_t)N * 64);

  const int lane = threadIdx.x & 31;
  const int half = lane >> 4;                // 0: K pair {0,1}; 1: K pair {2,3}
  const int l15  = lane & 15;

  int arow = m0 + l15;
  if (arow >= N) arow = N - 1;               // clamp loads; stores are guarded

  v8f acc0 = {}, acc1 = {}, acc2 = {}, acc3 = {};

  const float* __restrict__ fa = feat + (size_t)arow * 64;

#pragma unroll
  for (int k0 = 0; k0 < 64; k0 += 4) {
    const int ka = k0 + 2 * half;            // this lane's K pair
    v2f a;
    a.x = fa[ka];                            // contiguous pair -> global_load_b64
    a.y = fa[ka + 1];

    const float* __restrict__ w0 = W + (size_t)ka * 64;        // row K=ka
    const float* __restrict__ w1 = W + (size_t)(ka + 1) * 64;  // row K=ka+1

    v2f b0, b1, b2, b3;
    b0.x = w0[l15 +  0]; b0.y = w1[l15 +  0];
    b1.x = w0[l15 + 16]; b1.y = w1[l15 + 16];
    b2.x = w0[l15 + 32]; b2.y = w1[l15 + 32];
    b3.x = w0[l15 + 48]; b3.y = w1[l15 + 48];

    acc0 = __builtin_amdgcn_wmma_f32_16x16x4_f32(false, a, false, b0,
                                                 (short)0, acc0, false, false);
    acc1 = __builtin_amdgcn_wmma_f32_16x16x4_f32(false, a, false, b1,
                                                 (short)0, acc1, false, false);
    acc2 = __builtin_amdgcn_wmma_f32_16x16x4_f32(false, a, false, b2,
                                                 (short)0, acc2, false, false);
    acc3 = __builtin_amdgcn_wmma_f32_16x16x4_f32(false, a, false, b3,
                                                 (short)0, acc3, false, false);
  }

  // C/D layout: VGPR j -> row (m0 + j + 8*half), col l15 (+16/32/48 per tile)
#pragma unroll
  for (int j = 0; j < 8; ++j) {
    const int row = m0 + j + 8 * half;
    if (row < N) {
      float* __restrict__ drow = dst + (size_t)row * 64 + l15;
      drow[0]  = acc0[j];
      drow[16] = acc1[j];
      drow[32] = acc2[j];
      drow[48] = acc3[j];
    }
  }
}

// ---------------------------------------------------------------------------
// Kernel 2: per-edge gather from L2-resident proj, scale, scatter-add.
// 16 threads per edge, float4 chunk each -> coalesced 256B gathers,
// 4x global_atomic_add_f32 per thread.
// ---------------------------------------------------------------------------
__global__ void rgcn_edge_scatter(const float* __restrict__ proj,   // [R,N,64]
                                  const int* __restrict__ esrc,
                                  const int* __restrict__ edst,
                                  const int* __restrict__ etype,
                                  const float* __restrict__ att,    // [E]
                                  const float* __restrict__ bias,   // [64]
                                  float* __restrict__ out,          // [N,64]
                                  int E, int N) {
  const long long gid = (long long)blockIdx.x * blockDim.x + threadIdx.x;
  const int e = (int)(gid >> 4);
  const int q = (int)(gid & 15);
  if (e >= E) return;

  const int   s = esrc[e];
  const int   d = edst[e];
  const int   r = etype[e];
  const float a = att[e];

  const float4 p = *(const float4*)(proj +
      ((size_t)r * (size_t)N + (size_t)s) * 64 + q * 4);
  const float4 b = *(const float4*)(bias + q * 4);

  float* __restrict__ o = out + (size_t)d * 64 + q * 4;
  unsafeAtomicAdd(o + 0, (p.x + b.x) * a);
  unsafeAtomicAdd(o + 1, (p.y + b.y) * a);
  unsafeAtomicAdd(o + 2, (p.z + b.z) * a);
  unsafeAtomicAdd(o + 3, (p.w + b.w) * a);
}

extern "C" void kernel_launch(void* const* d_in, const int* in_sizes, int n_in,
                              void* d_out, int out_size, void* d_ws, size_t ws_size,
                              hipStream_t stream) {
  const float* feat   = (const float*)d_in[0];
  const int*   esrc   = (const int*)  d_in[1];
  const int*   edst   = (const int*)  d_in[2];
  const int*   etype  = (const int*)  d_in[3];
  const float* att    = (const float*)d_in[4];
  const float* weight = (const float*)d_in[5];
  const float* bias   = (const float*)d_in[6];
  const float* loop_w = (const float*)d_in[7];
  float*       out    = (float*)d_out;
  float*       proj   = (float*)d_ws;      // [R, N, 64] f32 (102.4 MB; L2-resident)

  const int N = in_sizes[0] / 64;
  const int E = in_sizes[1];
  const int R = in_sizes[5] / (64 * 64);

  // --- Stage 1: WMMA GEMMs (R relation projections + self-loop into d_out) ---
  const int n_mblocks   = (N + 15) / 16;
  const int total_waves = n_mblocks * (R + 1);
  const int waves_pb    = 8;                        // 256 threads = 8 wave32
  const int blocks1     = (total_waves + waves_pb - 1) / waves_pb;
  rgcn_proj_wmma<<<blocks1, waves_pb * 32, 0, stream>>>(
      feat, weight, loop_w, proj, out, N, R, n_mblocks);

  // --- Stage 2: edge gather/scale/scatter (atomic accumulate into d_out) ---
  const long long tthreads = (long long)E * 16;
  const int       tpb      = 256;
  const long long blocks2  = (tthreads + tpb - 1) / tpb;
  rgcn_edge_scatter<<<(unsigned)blocks2, tpb, 0, stream>>>(
      proj, esrc, edst, etype, att, bias, out, E, N);
}